// AdjustNet_89867895701799
// MI455X (gfx1250) — compile-verified
//
#include <hip/hip_runtime.h>
#include <math.h>

typedef float v2f __attribute__((ext_vector_type(2)));
typedef float v8f __attribute__((ext_vector_type(8)));

#define B_  32
#define L_  512
#define C_  256
#define N_  16
#define L4_ 128   // L/4

// D(16x16,f32) = A(16x4,f32) x B(4x16,f32) + C
// Layout (ISA 7.12.2): A: lane = M (0..15 twice), VGPR j holds K = 2*(lane>>4)+j
//                      B: lane = N, VGPR j holds K = 2*(lane>>4)+j
//                      C/D: lane = N, VGPR v holds M = v + 8*(lane>>4)
static __device__ __forceinline__ v8f wmma4(v2f a, v2f b, v8f c) {
  return __builtin_amdgcn_wmma_f32_16x16x4_f32(
      /*neg_a=*/false, a, /*neg_b=*/false, b,
      /*c_mod=*/(short)0, c, /*reuse_a=*/false, /*reuse_b=*/false);
}

static __device__ __forceinline__ float gelu_exact(float x) {
  return 0.5f * x * (1.0f + erff(x * 0.70710678118654752f));
}

// ---------------------------------------------------------------------------
// K1: t[m,0:128] = gelu(LN(query^T[m,:] @ w1 + b1)) for m in [0, B*C)
//     query^T[m,k] = query[(b*L + k)*C + c], m = b*C + c
// grid 512 blocks x 256 thr (8 waves); each wave one 16x16 N-tile (N=128 total)
// ---------------------------------------------------------------------------
__global__ __launch_bounds__(256) void adjnet_k1_sample_in(
    const float* __restrict__ query, const float* __restrict__ w1,
    const float* __restrict__ b1, const float* __restrict__ g1,
    const float* __restrict__ bt1, float* __restrict__ t_out) {
  __shared__ __align__(16) float As[L_ * 16];   // As[k*16 + r]  (32 KB)
  __shared__ float Tv[16 * 132];                // tile values, padded rows
  __shared__ float mu[16], rs[16];

  const int tid = threadIdx.x;
  const int m0  = blockIdx.x * 16;
  const int b   = m0 >> 8;      // c-block stays inside one batch (16 | 256)
  const int c0  = m0 & 255;

  // stage transposed A tile into LDS (float4 along c)
  {
    const int rq = (tid & 3) * 4;
    const int kk = tid >> 2;                    // 0..63
    for (int i = 0; i < 8; ++i) {
      const int k = kk + 64 * i;
      const float4 v =
          *(const float4*)(query + ((size_t)(b * L_ + k)) * C_ + c0 + rq);
      *(float4*)(As + k * 16 + rq) = v;
    }
  }
  __syncthreads();

  const int lane = tid & 31, wave = tid >> 5;
  const int half = lane >> 4, mr = lane & 15;
  const int kb   = 2 * half;
  const int ncol = 16 * wave + mr;

  v8f acc = {};
#pragma unroll 8
  for (int k0 = 0; k0 < L_; k0 += 4) {
    v2f a, bf;
    a.x  = As[(k0 + kb) * 16 + mr];
    a.y  = As[(k0 + kb + 1) * 16 + mr];
    bf.x = w1[(size_t)(k0 + kb) * L4_ + ncol];
    bf.y = w1[(size_t)(k0 + kb + 1) * L4_ + ncol];
    acc  = wmma4(a, bf, acc);
  }
  {
    const float bb = b1[ncol];
#pragma unroll
    for (int v = 0; v < 8; ++v) Tv[(v + 8 * half) * 132 + ncol] = acc[v] + bb;
  }
  __syncthreads();

  if (tid < 16) {
    float s = 0.f, s2 = 0.f;
    for (int kx = 0; kx < L4_; ++kx) {
      const float h = Tv[tid * 132 + kx];
      s += h; s2 += h * h;
    }
    const float m = s * (1.0f / L4_);
    mu[tid] = m;
    rs[tid] = rsqrtf(s2 * (1.0f / L4_) - m * m + 1e-12f);
  }
  __syncthreads();

  for (int idx = tid; idx < 16 * L4_; idx += 256) {
    const int r = idx >> 7, kx = idx & 127;
    const float h = (Tv[r * 132 + kx] - mu[r]) * rs[r] * g1[kx] + bt1[kx];
    t_out[(size_t)(m0 + r) * L4_ + kx] = gelu_exact(h);
  }
}

// ---------------------------------------------------------------------------
// K2: q2[b,l,c] = t[m,:] @ w2[:,l] + b2[l], m = b*C + c  (store transposed)
// grid (512, 4) x 256 thr; wave covers 16 cols; K = 128
// ---------------------------------------------------------------------------
__global__ __launch_bounds__(256) void adjnet_k2_sample_out(
    const float* __restrict__ t_in, const float* __restrict__ w2,
    const float* __restrict__ b2, float* __restrict__ q2) {
  __shared__ float As[16 * 132];
  const int tid = threadIdx.x;
  const int m0  = blockIdx.x * 16;
  const int nb  = blockIdx.y * 128;

  for (int idx = tid; idx < 16 * 128; idx += 256) {
    const int r = idx >> 7, kx = idx & 127;
    As[r * 132 + kx] = t_in[(size_t)(m0 + r) * L4_ + kx];
  }
  __syncthreads();

  const int lane = tid & 31, wave = tid >> 5;
  const int half = lane >> 4, mr = lane & 15;
  const int kb = 2 * half;
  const int n0 = nb + 16 * wave;

  v8f acc = {};
#pragma unroll 8
  for (int k0 = 0; k0 < L4_; k0 += 4) {
    v2f a, bf;
    a.x  = As[mr * 132 + k0 + kb];
    a.y  = As[mr * 132 + k0 + kb + 1];
    bf.x = w2[(size_t)(k0 + kb) * L_ + n0 + mr];
    bf.y = w2[(size_t)(k0 + kb + 1) * L_ + n0 + mr];
    acc  = wmma4(a, bf, acc);
  }
  const int   ncol = n0 + mr;
  const float bb   = b2[ncol];
  const int   b    = m0 >> 8;
  const int   cc0  = m0 & 255;
#pragma unroll
  for (int v = 0; v < 8; ++v) {
    q2[((size_t)(b * L_ + ncol)) * C_ + cc0 + v + 8 * half] = acc[v] + bb;
  }
}

// ---------------------------------------------------------------------------
// K3: adj[m,0:32] = tanh(gelu(LN(q2[m,:] @ w3 + b3)) @ w4 + b4) * 0.02
// grid 1024 x 256 thr; each wave two 16x16 tiles for GEMM3 (N=256);
// GEMM4 (N=32) handled by waves 0,1 with A-fragments read from LDS.
// ---------------------------------------------------------------------------
__global__ __launch_bounds__(256) void adjnet_k3_adjust(
    const float* __restrict__ q2, const float* __restrict__ w3,
    const float* __restrict__ b3, const float* __restrict__ g3,
    const float* __restrict__ bt3, const float* __restrict__ w4,
    const float* __restrict__ b4, float* __restrict__ adj) {
  __shared__ float As[16 * 264];
  __shared__ float Hh[16 * 264];
  __shared__ float mu[16], rs[16];
  const int tid = threadIdx.x;
  const int m0  = blockIdx.x * 16;

  for (int idx = tid; idx < 16 * C_; idx += 256) {
    const int r = idx >> 8, kx = idx & 255;
    As[r * 264 + kx] = q2[(size_t)(m0 + r) * C_ + kx];
  }
  __syncthreads();

  const int lane = tid & 31, wave = tid >> 5;
  const int half = lane >> 4, mr = lane & 15;
  const int kb = 2 * half;

  for (int tn = 0; tn < 2; ++tn) {
    const int n0 = 32 * wave + 16 * tn;
    v8f acc = {};
#pragma unroll 8
    for (int k0 = 0; k0 < C_; k0 += 4) {
      v2f a, bf;
      a.x  = As[mr * 264 + k0 + kb];
      a.y  = As[mr * 264 + k0 + kb + 1];
      bf.x = w3[(size_t)(k0 + kb) * C_ + n0 + mr];
      bf.y = w3[(size_t)(k0 + kb + 1) * C_ + n0 + mr];
      acc  = wmma4(a, bf, acc);
    }
    const int   ncol = n0 + mr;
    const float bb   = b3[ncol];
#pragma unroll
    for (int v = 0; v < 8; ++v) Hh[(v + 8 * half) * 264 + ncol] = acc[v] + bb;
  }
  __syncthreads();

  if (tid < 16) {
    float s = 0.f, s2 = 0.f;
    for (int kx = 0; kx < C_; ++kx) {
      const float h = Hh[tid * 264 + kx];
      s += h; s2 += h * h;
    }
    const float m = s * (1.0f / C_);
    mu[tid] = m;
    rs[tid] = rsqrtf(s2 * (1.0f / C_) - m * m + 1e-12f);
  }
  __syncthreads();

  for (int idx = tid; idx < 16 * C_; idx += 256) {
    const int r = idx >> 8, kx = idx & 255;
    const float h = (Hh[r * 264 + kx] - mu[r]) * rs[r] * g3[kx] + bt3[kx];
    Hh[r * 264 + kx] = gelu_exact(h);
  }
  __syncthreads();

  if (wave < 2) {  // wave-uniform branch: EXEC all-1s inside the wave (WMMA ok)
    const int n0 = 16 * wave;
    v8f acc = {};
#pragma unroll 8
    for (int k0 = 0; k0 < C_; k0 += 4) {
      v2f a, bf;
      a.x  = Hh[mr * 264 + k0 + kb];
      a.y  = Hh[mr * 264 + k0 + kb + 1];
      bf.x = w4[(size_t)(k0 + kb) * 32 + n0 + mr];
      bf.y = w4[(size_t)(k0 + kb + 1) * 32 + n0 + mr];
      acc  = wmma4(a, bf, acc);
    }
    const int   ncol = n0 + mr;
    const float bb   = b4[ncol];
#pragma unroll
    for (int v = 0; v < 8; ++v) {
      adj[(size_t)(m0 + v + 8 * half) * 32 + ncol] = tanhf(acc[v] + bb) * 0.02f;
    }
  }
}

// ---------------------------------------------------------------------------
// K4: sampling (bandwidth-bound): pos, offset passthrough, bilinear taps
// grid B*L x 256 thr; threads stream 16 coalesced 1KB rows of `out`
// ---------------------------------------------------------------------------
__global__ __launch_bounds__(256) void adjnet_k4_sample(
    const float* __restrict__ x, const float* __restrict__ offset,
    const float* __restrict__ adj, float* __restrict__ out_main,
    float* __restrict__ out_off, float* __restrict__ out_pos) {
  __shared__ float w0s[16], w1s[16];
  __shared__ int   c0s[16], c1s[16];
  const int tid = threadIdx.x;
  const int m   = blockIdx.x;   // b*L + l
  const int l   = m & (L_ - 1);
  const int b   = m >> 9;

  if (tid < 16) {
    const int   n  = tid;
    const float ox = offset[(size_t)m * 32 + 2 * n];
    const float oy = offset[(size_t)m * 32 + 2 * n + 1];
    const float ax = adj[(size_t)m * 32 + 2 * n];
    const float ay = adj[(size_t)m * 32 + 2 * n + 1];
    const float refy =
        ((float)n + 0.5f) * 2.0f * ((float)l + 1e-9f) / (float)(N_ * L_) - 1.0f;
    const float px = tanhf(ox + ax);
    const float py = tanhf(oy + ay + refy);
    out_pos[(size_t)m * 32 + 2 * n]     = px;
    out_pos[(size_t)m * 32 + 2 * n + 1] = py;
    const float pix = (py + 1.0f) * 0.5f * (float)(L_ - 1);
    const float i0f = floorf(pix);
    const float fr  = pix - i0f;
    const int   i0  = (int)i0f;
    const int   i1  = i0 + 1;
    const bool  v0  = (i0 >= 0) && (i0 < L_) && (i0 <= l);
    const bool  v1  = (i1 >= 0) && (i1 < L_) && (i1 <= l);
    c0s[n] = min(max(i0, 0), L_ - 1);
    c1s[n] = min(max(i1, 0), L_ - 1);
    w0s[n] = v0 ? (1.0f - fr) : 0.0f;
    w1s[n] = v1 ? fr : 0.0f;
  }
  if (tid < 32) out_off[(size_t)m * 32 + tid] = offset[(size_t)m * 32 + tid];
  __syncthreads();

  const float* xb = x + (size_t)b * L_ * C_;
  float*       ob = out_main + (size_t)m * N_ * C_;
#pragma unroll
  for (int n = 0; n < N_; ++n) {
    ob[n * C_ + tid] = w0s[n] * xb[(size_t)c0s[n] * C_ + tid] +
                       w1s[n] * xb[(size_t)c1s[n] * C_ + tid];
  }
}

// ---------------------------------------------------------------------------
extern "C" void kernel_launch(void* const* d_in, const int* in_sizes, int n_in,
                              void* d_out, int out_size, void* d_ws,
                              size_t ws_size, hipStream_t stream) {
  (void)in_sizes; (void)n_in; (void)out_size; (void)ws_size;
  const float* query = (const float*)d_in[0];
  const float* x     = (const float*)d_in[1];
  const float* off   = (const float*)d_in[2];
  const float* w1    = (const float*)d_in[3];
  const float* b1    = (const float*)d_in[4];
  const float* g1    = (const float*)d_in[5];
  const float* bt1   = (const float*)d_in[6];
  const float* w2    = (const float*)d_in[7];
  const float* b2    = (const float*)d_in[8];
  const float* w3    = (const float*)d_in[9];
  const float* b3    = (const float*)d_in[10];
  const float* g3    = (const float*)d_in[11];
  const float* bt3   = (const float*)d_in[12];
  const float* w4    = (const float*)d_in[13];
  const float* b4    = (const float*)d_in[14];

  float* out     = (float*)d_out;
  float* out_off = out + (size_t)B_ * L_ * N_ * C_;      // 67,108,864
  float* out_pos = out_off + (size_t)B_ * L_ * 2 * N_;   // +524,288

  float* ws     = (float*)d_ws;
  float* t_ws   = ws;                       // 8192*128 floats (4 MB)
  float* q2_ws  = ws + (size_t)8192 * 128;  // 16384*256 floats (16 MB)
  float* adj_ws = ws;                       // overlays t (t dead after K2)

  adjnet_k1_sample_in<<<512, 256, 0, stream>>>(query, w1, b1, g1, bt1, t_ws);
  adjnet_k2_sample_out<<<dim3(512, 4), 256, 0, stream>>>(t_ws, w2, b2, q2_ws);
  adjnet_k3_adjust<<<1024, 256, 0, stream>>>(q2_ws, w3, b3, g3, bt3, w4, b4,
                                             adj_ws);
  adjnet_k4_sample<<<B_ * L_, 256, 0, stream>>>(x, off, adj_ws, out, out_off,
                                                out_pos);
}